// ElectraClassificationHeadSentiment_69965017252051
// MI455X (gfx1250) — compile-verified
//
#include <hip/hip_runtime.h>
#include <math.h>

// Problem constants (match reference setup_inputs()).
constexpr int B = 64;
constexpr int S = 2048;
constexpr int H = 1024;
constexpr int L = 3;

// Sequence-split factor for the bandwidth-bound pooling phase:
// 16 chunks x 64 batches = 1024 blocks in flight -> enough independent
// request streams to approach the 23.3 TB/s HBM roofline.
constexpr int SPLIT = 16;
constexpr int CHUNK = S / SPLIT;  // 128 rows per chunk

typedef __attribute__((ext_vector_type(2))) float v2f;
typedef __attribute__((ext_vector_type(4))) float v4f;
typedef __attribute__((ext_vector_type(8))) float v8f;

// ---------------------------------------------------------------------------
// Kernel 1a: partial masked sums.
// Grid (SPLIT, B); 256 threads; each thread owns 4 contiguous H-columns.
// 128-bit non-temporal loads (features is streamed exactly once -> NT hint
// keeps it from displacing L2-resident weights).
// ---------------------------------------------------------------------------
__global__ void pool_partial(const float* __restrict__ features,
                             const int* __restrict__ ner_ids,
                             float* __restrict__ partial) {
  const int chunk = blockIdx.x;
  const int b     = blockIdx.y;
  const int start = ner_ids[2 * b + 0];
  const int end   = ner_ids[2 * b + 1];

  const int lo = max(start, chunk * CHUNK);
  const int hi = min(end, chunk * CHUNK + (CHUNK - 1));

  const int tid = threadIdx.x;  // 0..255
  const int h4  = tid * 4;
  const float* base = features + (size_t)b * S * H + h4;

  v4f acc = {0.f, 0.f, 0.f, 0.f};
  for (int s = lo; s <= hi; ++s) {
    v4f v = __builtin_nontemporal_load((const v4f*)(base + (size_t)s * H));
    acc += v;
  }
  // Always write (zeros when the chunk misses the span) so the reduce kernel
  // never sees poisoned workspace.
  *(v4f*)(partial + ((size_t)b * SPLIT + chunk) * H + h4) = acc;
}

// ---------------------------------------------------------------------------
// Kernel 1b: deterministic fixed-order reduction of the SPLIT partials + mean.
// ---------------------------------------------------------------------------
__global__ void pool_reduce(const float* __restrict__ partial,
                            const int* __restrict__ ner_ids,
                            float* __restrict__ pooled) {
  const int b   = blockIdx.x;
  const int tid = threadIdx.x;
  const int h4  = tid * 4;

  v4f acc = {0.f, 0.f, 0.f, 0.f};
#pragma unroll
  for (int c = 0; c < SPLIT; ++c) {
    acc += *(const v4f*)(partial + ((size_t)b * SPLIT + c) * H + h4);
  }
  const int start = ner_ids[2 * b + 0];
  const int end   = ner_ids[2 * b + 1];
  const float inv = 1.0f / (float)(end - start + 1);
  *(v4f*)(pooled + (size_t)b * H + h4) = acc * inv;
}

// ---------------------------------------------------------------------------
// Kernel 2: x = gelu(pooled @ W_dense + b_dense) via V_WMMA_F32_16X16X4_F32.
// Wave32: one 16x16 output tile per wave. 4 M-tiles x 64 N-tiles = 256 waves
// = 32 blocks x 8 waves. K loop: 1024/4 = 256 WMMA ops per wave.
//
// Fragment layouts (ISA 7.12.2):
//   A (16x4 f32, 2 VGPRs): lanes 0-15 -> M=lane, K={0,1}; lanes 16-31 -> K={2,3}
//   B (4x16 f32, 2 VGPRs): lanes 0-15 -> N=lane, K={0,1}; lanes 16-31 -> K={2,3}
//   C/D (16x16 f32, 8 VGPRs): vgpr r -> M=r (lanes 0-15) / M=r+8 (lanes 16-31),
//                             N = lane % 16
// ---------------------------------------------------------------------------
__device__ __forceinline__ float gelu_exact(float x) {
  // 0.5 * x * (1 + erf(x / sqrt(2)))  -- matches approximate=False
  return 0.5f * x * (1.0f + erff(x * 0.70710678118654752440f));
}

__global__ void dense_gelu_wmma(const float* __restrict__ pooled,
                                const float* __restrict__ Wd,
                                const float* __restrict__ bd,
                                float* __restrict__ xout) {
  const int tid  = threadIdx.x;
  const int lane = tid & 31;
  const int wave = tid >> 5;
  const int gw   = blockIdx.x * (blockDim.x >> 5) + wave;  // 0..255

  const int mt  = gw & 3;    // M tile: 0..3   (M = 64)
  const int nt  = gw >> 2;   // N tile: 0..63  (N = 1024)
  const int row = lane & 15; // M-row (for A) / N-col (for B)
  const int kh  = lane >> 4; // K half-select: 0 -> K{0,1}, 1 -> K{2,3}

  const float* Abase = pooled + (size_t)(mt * 16 + row) * H;  // row of pooled
  const float* Bbase = Wd + nt * 16 + row;                    // column of W_dense

  v8f c = {};
#pragma unroll 4
  for (int k0 = 0; k0 < H; k0 += 4) {
    const int ka = k0 + kh * 2;
    v2f a, bfrag;
    a.x     = Abase[ka];
    a.y     = Abase[ka + 1];
    bfrag.x = Bbase[(size_t)ka * H];
    bfrag.y = Bbase[(size_t)(ka + 1) * H];
    // 8 args: (neg_a, A, neg_b, B, c_mod, C, reuse_a, reuse_b)
    c = __builtin_amdgcn_wmma_f32_16x16x4_f32(
        /*neg_a=*/false, a, /*neg_b=*/false, bfrag,
        /*c_mod=*/(short)0, c, /*reuse_a=*/false, /*reuse_b=*/false);
  }

  const int col   = nt * 16 + row;
  const float bia = bd[col];
#pragma unroll
  for (int r = 0; r < 8; ++r) {
    const int m = mt * 16 + r + kh * 8;
    xout[(size_t)m * H + col] = gelu_exact(c[r] + bia);
  }
}

// ---------------------------------------------------------------------------
// Kernel 3: out = x @ W_out + b_out   (64 x 1024 x 3; trivial reduction)
// One block per batch row; 256 threads reduce the 3 dot products in LDS.
// ---------------------------------------------------------------------------
__global__ void out_proj(const float* __restrict__ x,
                         const float* __restrict__ Wo,
                         const float* __restrict__ bo,
                         float* __restrict__ out) {
  const int b   = blockIdx.x;
  const int tid = threadIdx.x;  // 0..255

  float a0 = 0.f, a1 = 0.f, a2 = 0.f;
  for (int h = tid; h < H; h += 256) {
    const float xv = x[(size_t)b * H + h];
    a0 += xv * Wo[h * L + 0];
    a1 += xv * Wo[h * L + 1];
    a2 += xv * Wo[h * L + 2];
  }

  __shared__ float sm[3][256];
  sm[0][tid] = a0; sm[1][tid] = a1; sm[2][tid] = a2;
  __syncthreads();
  for (int off = 128; off > 0; off >>= 1) {
    if (tid < off) {
      sm[0][tid] += sm[0][tid + off];
      sm[1][tid] += sm[1][tid + off];
      sm[2][tid] += sm[2][tid + off];
    }
    __syncthreads();
  }
  if (tid == 0) {
    out[b * L + 0] = sm[0][0] + bo[0];
    out[b * L + 1] = sm[1][0] + bo[1];
    out[b * L + 2] = sm[2][0] + bo[2];
  }
}

// ---------------------------------------------------------------------------
// Launch
// ---------------------------------------------------------------------------
extern "C" void kernel_launch(void* const* d_in, const int* in_sizes, int n_in,
                              void* d_out, int out_size, void* d_ws, size_t ws_size,
                              hipStream_t stream) {
  const float* features = (const float*)d_in[0];  // (B, S, H) fp32
  const int*   ner_ids  = (const int*)d_in[1];    // (B, 2)
  const float* W_dense  = (const float*)d_in[2];  // (H, H)
  const float* b_dense  = (const float*)d_in[3];  // (H,)
  const float* W_out    = (const float*)d_in[4];  // (H, L)
  const float* b_out    = (const float*)d_in[5];  // (L,)
  float*       out      = (float*)d_out;          // (B, L)

  // Workspace layout: partial (B*SPLIT*H) | pooled (B*H) | x (B*H), fp32.
  // Everything is fully written before being read (pool_partial writes zeros
  // for empty chunks), so the 0xAA poison never leaks.
  float* partial = (float*)d_ws;                       // 4 MB
  float* pooled  = partial + (size_t)B * SPLIT * H;    // 256 KB
  float* xbuf    = pooled + (size_t)B * H;             // 256 KB

  dim3 pgrid(SPLIT, B);
  pool_partial<<<pgrid, 256, 0, stream>>>(features, ner_ids, partial);
  pool_reduce<<<B, 256, 0, stream>>>(partial, ner_ids, pooled);
  dense_gelu_wmma<<<(B / 16) * (H / 16) / 8, 256, 0, stream>>>(pooled, W_dense,
                                                              b_dense, xbuf);
  out_proj<<<B, 256, 0, stream>>>(xbuf, W_out, b_out, out);
}